// RFNTDetector_v2_61838939128076
// MI455X (gfx1250) — compile-verified
//
#include <hip/hip_runtime.h>
#include <hip/hip_bf16.h>

// Problem constants from the reference
#define B 128
#define D 768
#define C 2
#define BN_EPS 1e-5f

typedef __attribute__((ext_vector_type(2))) float v2f;
typedef __attribute__((ext_vector_type(8))) float v8f;

// ---------------------------------------------------------------------------
// Kernel 1: per-sample stats + rank-1 max-pool collapse.
//   pooled[b,d] = x1c * (x1c >= 0 ? max(x2c) : min(x2c)) / (D-1)
// One block per sample b, 256 threads reduce D=768 elements.
// ---------------------------------------------------------------------------
__global__ void rfnt_rowstats_pool(const float* __restrict__ x1,
                                   const float* __restrict__ x2,
                                   float* __restrict__ pooled) {
    const int b   = blockIdx.x;
    const int tid = threadIdx.x;
    const float* r1 = x1 + b * D;
    const float* r2 = x2 + b * D;

    __shared__ float s_s1[256], s_s2[256], s_mx[256], s_mn[256];

    float s1 = 0.f, s2 = 0.f, mx = -3.4e38f, mn = 3.4e38f;
    for (int i = tid; i < D; i += 256) {
        float a = r1[i];
        float c = r2[i];
        s1 += a;
        s2 += c;
        mx = fmaxf(mx, c);
        mn = fminf(mn, c);
    }
    s_s1[tid] = s1; s_s2[tid] = s2; s_mx[tid] = mx; s_mn[tid] = mn;
    __syncthreads();
    for (int s = 128; s > 0; s >>= 1) {
        if (tid < s) {
            s_s1[tid] += s_s1[tid + s];
            s_s2[tid] += s_s2[tid + s];
            s_mx[tid] = fmaxf(s_mx[tid], s_mx[tid + s]);
            s_mn[tid] = fminf(s_mn[tid], s_mn[tid + s]);
        }
        __syncthreads();
    }
    const float m1  = s_s1[0] * (1.0f / D);
    const float m2  = s_s2[0] * (1.0f / D);
    const float hiC = s_mx[0] - m2;   // max of centered x2
    const float loC = s_mn[0] - m2;   // min of centered x2
    const float inv = 1.0f / (float)(D - 1);

    for (int i = tid; i < D; i += 256) {
        float x1c = r1[i] - m1;
        pooled[b * D + i] = x1c * (x1c >= 0.f ? hiC : loC) * inv;
    }
}

// ---------------------------------------------------------------------------
// Kernel 2: batch-norm statistics over the batch axis, folded:
//   scale[d] = gamma[d] * rsqrt(var[d] + eps)
//   shift[d] = beta[d]  - mu[d] * scale[d]
// One thread per feature column d (coalesced: consecutive d adjacent).
// ---------------------------------------------------------------------------
__global__ void rfnt_bn_stats(const float* __restrict__ pooled,
                              const float* __restrict__ gamma,
                              const float* __restrict__ beta,
                              float* __restrict__ scale,
                              float* __restrict__ shift) {
    const int d = blockIdx.x * blockDim.x + threadIdx.x;
    if (d >= D) return;
    float sum = 0.f, sq = 0.f;
    for (int b = 0; b < B; ++b) {
        float v = pooled[b * D + d];
        sum += v;
        sq  += v * v;
    }
    const float mu  = sum * (1.0f / B);
    const float var = sq * (1.0f / B) - mu * mu;   // biased, matches jnp.var
    const float sc  = rsqrtf(var + BN_EPS) * gamma[d];
    scale[d] = sc;
    shift[d] = beta[d] - mu * sc;
}

// ---------------------------------------------------------------------------
// Kernel 3: classifier GEMM via V_WMMA_F32_16X16X4_F32 (exact f32 path).
//   out[m,n] = sum_k (pooled[m,k]*scale[k] + shift[k]) * Wpad[n,k] + bias[n]
// 1 block, 256 threads = 8 waves. Wave w owns M-rows [16w,16w+16).
// N padded 2->16 with a zero-filled W tile in LDS; K=768 in 192 steps of 4.
// A frag (16x4): lane l -> M=l%16, Ks {2*(l/16), 2*(l/16)+1}.
// B frag (4x16): lane l -> N=l%16, same K pair. D: VGPR v -> row v+8*(l/16).
// EXEC stays all-1s through the WMMA loop (required by ISA).
// ---------------------------------------------------------------------------
__global__ void rfnt_classifier_wmma(const float* __restrict__ pooled,
                                     const float* __restrict__ scale,
                                     const float* __restrict__ shift,
                                     const float* __restrict__ W,
                                     const float* __restrict__ bias,
                                     float* __restrict__ out) {
    __shared__ float s_W[16 * D];     // W padded to 16 rows (zeros for n>=2)
    __shared__ float s_scale[D];
    __shared__ float s_shift[D];

    const int tid = threadIdx.x;

    // Cooperative LDS fill (uniform trip counts; 16*768 % 256 == 0)
    for (int i = tid; i < 16 * D; i += 256) s_W[i] = 0.f;
    for (int i = tid; i < D; i += 256) {
        s_scale[i] = scale[i];
        s_shift[i] = shift[i];
    }
    __syncthreads();
    for (int i = tid; i < C * D; i += 256) s_W[i] = W[i];
    __syncthreads();

    const int wave = tid >> 5;
    const int lane = tid & 31;
    const int m    = lane & 15;       // M index for A, N index for B
    const int hi   = lane >> 4;
    const int row  = 16 * wave + m;
    const float* prow = pooled + row * D;
    const float* wrow = s_W + m * D;  // B-fragment row (N = lane%16)

    v8f acc = {};
    for (int k0 = 0; k0 < D; k0 += 4) {
        const int kx = k0 + 2 * hi;
        v2f a, bb;
        a.x  = fmaf(prow[kx],     s_scale[kx],     s_shift[kx]);
        a.y  = fmaf(prow[kx + 1], s_scale[kx + 1], s_shift[kx + 1]);
        bb.x = wrow[kx];
        bb.y = wrow[kx + 1];
        acc = __builtin_amdgcn_wmma_f32_16x16x4_f32(
            /*neg_a=*/false, a, /*neg_b=*/false, bb,
            /*c_mod=*/(short)0, acc, /*reuse_a=*/false, /*reuse_b=*/false);
    }

    // D-matrix layout: VGPR v -> (row v + 8*hi, col lane%16). Only n<2 valid.
    const int n = m;
    if (n < C) {
        const float bn = bias[n];
#pragma unroll
        for (int v = 0; v < 8; ++v) {
            const int mrow = v + 8 * hi;
            out[(16 * wave + mrow) * C + n] = acc[v] + bn;
        }
    }
}

// ---------------------------------------------------------------------------
extern "C" void kernel_launch(void* const* d_in, const int* in_sizes, int n_in,
                              void* d_out, int out_size, void* d_ws, size_t ws_size,
                              hipStream_t stream) {
    const float* feat_orig  = (const float*)d_in[0];   // [B, D]
    const float* feat_trans = (const float*)d_in[1];   // [B, D]
    const float* bn_gamma   = (const float*)d_in[2];   // [D]
    const float* bn_beta    = (const float*)d_in[3];   // [D]
    const float* W          = (const float*)d_in[4];   // [C, D]
    const float* bias       = (const float*)d_in[5];   // [C]
    float* out = (float*)d_out;                        // [B, C]

    float* pooled = (float*)d_ws;                      // B*D floats
    float* scale  = pooled + B * D;                    // D floats
    float* shift  = scale + D;                         // D floats

    rfnt_rowstats_pool<<<B, 256, 0, stream>>>(feat_orig, feat_trans, pooled);
    rfnt_bn_stats<<<(D + 255) / 256, 256, 0, stream>>>(pooled, bn_gamma, bn_beta,
                                                       scale, shift);
    rfnt_classifier_wmma<<<1, 256, 0, stream>>>(pooled, scale, shift, W, bias, out);
}